// UniEmbedding__60859686584405
// MI455X (gfx1250) — compile-verified
//
#include <hip/hip_runtime.h>
#include <cstdint>

// Problem constants (match reference)
constexpr int T_BS = 8;
constexpr int T_I  = 1024;
constexpr int T_J  = 64;
constexpr int T_K  = 256;
constexpr int NUM_EMB = 512;
constexpr int INDEX_OFFSET = 150;
constexpr int NROWS = T_BS * T_I;           // 8192 (b,i) rows
constexpr int NBLOCKS = 1024;               // fixed: 8 rows per block
constexpr int ROWS_PER_BLOCK = NROWS / NBLOCKS;  // 8
constexpr float INV_SQRT_J = 0.125f;        // 1/sqrt(64)
constexpr float INV_2PI = 0.15915494309189535f;

// Generic pointer -> 32-bit LDS byte offset (addrspacecast to AS3, truncate)
__device__ __forceinline__ unsigned to_lds(const void* p) {
    return (unsigned)(uintptr_t)(__attribute__((address_space(3))) const void*)p;
}

// One async 16B global->LDS copy (GVS mode: saddr base + 32-bit voffset)
__device__ __forceinline__ void async_copy_b128(unsigned lds_byte, unsigned goff_byte,
                                                const void* sbase) {
    asm volatile("global_load_async_to_lds_b128 %0, %1, %2"
                 :
                 : "v"(lds_byte), "v"(goff_byte), "s"(sbase)
                 : "memory");
}

__device__ __forceinline__ void wait_async_zero() {
    asm volatile("s_wait_asynccnt 0" ::: "memory");
}

// ---------------------------------------------------------------------------
// Kernel 1: interleaved table EP[t][k] = { exp(emb[t][k]), phase[t][k]/(2*pi) }
// 512x256 float2 = 1 MB into workspace (one-time).
// ---------------------------------------------------------------------------
__global__ __launch_bounds__(256) void ep_table_kernel(const float* __restrict__ emb,
                                                       const float* __restrict__ phase,
                                                       float2* __restrict__ ep) {
    int idx = blockIdx.x * 256 + threadIdx.x;
    if (idx < NUM_EMB * T_K)
        ep[idx] = make_float2(__expf(emb[idx]), phase[idx] * INV_2PI);
}

// ---------------------------------------------------------------------------
// Kernel 2 (fast path): fused gather + cos-sum, interleaved tables.
// LDS: [ EP_pos 64x256 float2 (128 KB) | meta 8x64 float2 (4 KB) ]
//   meta[i*64+j] = { inputs[r_i, j]/(2*pi), bitcast(type[r_i, j]) }
// d/(2*pi) = x' * e_pos * e  +  (q + q_pos)   -> one v_fma + v_mul + v_add + v_cos
// ---------------------------------------------------------------------------
__global__ __launch_bounds__(256) void uniemb_fast(const float* __restrict__ inputs,
                                                   const int*   __restrict__ types,
                                                   const float2* __restrict__ EP,
                                                   float*       __restrict__ out) {
    extern __shared__ float2 smem2[];
    float2* sEP   = smem2;                 // T_J*T_K entries (128 KB)
    float2* sMeta = smem2 + T_J * T_K;     // ROWS_PER_BLOCK*T_J entries (4 KB)

    const int tid = threadIdx.x;           // 0..255, thread <-> k column

    // ---- Stage positional EP rows [150,214) into LDS via async DMA:
    // 64 rows * 256 * 8B = 128 KB = 8192 x 16B chunks; 32 chunks per thread.
    {
        const float2* gEP = EP + INDEX_OFFSET * T_K;
        const unsigned base = to_lds(sEP);
#pragma unroll
        for (int c = 0; c < 32; ++c) {
            const unsigned off = (unsigned)(c * 256 + tid) * 16u;
            async_copy_b128(base + off, off, gEP);
        }
    }

    // ---- Stage all 8 rows' (scaled input, type) meta while DMA is in flight.
#pragma unroll
    for (int v = 0; v < 2; ++v) {
        const int idx = v * 256 + tid;         // 0..511
        const int i   = idx >> 6;              // row slot 0..7
        const int j   = idx & (T_J - 1);
        const int r   = blockIdx.x + (i << 10);
        sMeta[idx] = make_float2(inputs[r * T_J + j] * INV_2PI,
                                 __int_as_float(types[r * T_J + j]));
    }

    wait_async_zero();     // this wave's async loads complete
    __syncthreads();       // single barrier: all LDS (DMA + meta) visible

    const int k = tid;
#pragma unroll
    for (int i = 0; i < ROWS_PER_BLOCK; ++i) {
        const int r = blockIdx.x + (i << 10);
        const float2* meta = sMeta + i * T_J;
        float acc = 0.0f;
#pragma unroll 4
        for (int j = 0; j < T_J; ++j) {
            const float2 m  = meta[j];                 // ds b64, uniform broadcast
            const int    t  = __float_as_int(m.y);
            const float2 g  = EP[t * T_K + k];         // global b64, coalesced, L2-resident
            const float2 lp = sEP[j * T_K + k];        // ds b64, conflict-free
            const float  d  = fmaf(m.x * lp.x, g.x, g.y + lp.y);
            acc += __builtin_amdgcn_cosf(d);           // v_cos_f32 (input in revolutions)
        }
        out[r * T_K + k] = acc * INV_SQRT_J;
    }
}

// ---------------------------------------------------------------------------
// Fallback (ws too small for the 1 MB table): exp inline from raw emb.
// LDS: [ E_pos 64x256 f32 | Ph_pos 64x256 f32 | row 64 f32 | type 64 i32 ]
// ---------------------------------------------------------------------------
__global__ __launch_bounds__(256) void uniemb_fallback(const float* __restrict__ inputs,
                                                       const int*   __restrict__ types,
                                                       const float* __restrict__ phase,
                                                       const float* __restrict__ emb,
                                                       float*       __restrict__ out) {
    extern __shared__ float smem[];
    float* sEpos = smem;                        // T_J*T_K
    float* sPpos = smem + T_J * T_K;            // T_J*T_K
    float* sRow  = smem + 2 * T_J * T_K;        // T_J
    int*   sType = (int*)(sRow + T_J);          // T_J

    const int tid = threadIdx.x;

    {
        const float* gE = emb   + INDEX_OFFSET * T_K;
        const float* gP = phase + INDEX_OFFSET * T_K;
        const unsigned eBase = to_lds(sEpos);
        const unsigned pBase = to_lds(sPpos);
#pragma unroll
        for (int c = 0; c < 16; ++c) {
            const unsigned off = (unsigned)(c * 256 + tid) * 16u;
            async_copy_b128(eBase + off, off, gE);
            async_copy_b128(pBase + off, off, gP);
        }
        wait_async_zero();
    }
    __syncthreads();
    for (int idx = tid; idx < T_J * T_K; idx += 256)
        sEpos[idx] = __expf(sEpos[idx]);        // raw emb rows -> exp in place
    __syncthreads();

    const int k = tid;
    for (int r = blockIdx.x; r < NROWS; r += gridDim.x) {
        if (tid < T_J) {
            sRow[tid]  = inputs[r * T_J + tid];
            sType[tid] = types[r * T_J + tid];
        }
        __syncthreads();
        float acc = 0.0f;
#pragma unroll 4
        for (int j = 0; j < T_J; ++j) {
            const int   t  = sType[j];
            const float x  = sRow[j];
            const float e  = __expf(emb[t * T_K + k]);
            const float p  = phase[t * T_K + k];
            const float ep = sEpos[j * T_K + k];
            const float pp = sPpos[j * T_K + k];
            const float d  = fmaf(x * ep, e, p + pp);
            acc += __builtin_amdgcn_cosf(d * INV_2PI);
        }
        out[r * T_K + k] = acc * INV_SQRT_J;
        __syncthreads();
    }
}

// ---------------------------------------------------------------------------
extern "C" void kernel_launch(void* const* d_in, const int* in_sizes, int n_in,
                              void* d_out, int out_size, void* d_ws, size_t ws_size,
                              hipStream_t stream) {
    const float* inputs = (const float*)d_in[0];   // [8,1024,64] f32
    const int*   types  = (const int*)  d_in[1];   // [8,1024,64] i32
    const float* emb    = (const float*)d_in[2];   // [512,256]   f32
    const float* phase  = (const float*)d_in[3];   // [512,256]   f32
    float*       out    = (float*)d_out;           // [8,1024,256] f32

    const size_t ep_bytes = (size_t)NUM_EMB * T_K * sizeof(float2);   // 1 MB

    if (ws_size >= ep_bytes) {
        float2* ep = (float2*)d_ws;
        ep_table_kernel<<<(NUM_EMB * T_K) / 256, 256, 0, stream>>>(emb, phase, ep);
        const size_t shmem = (size_t)(T_J * T_K + ROWS_PER_BLOCK * T_J) * sizeof(float2);
        uniemb_fast<<<NBLOCKS, 256, shmem, stream>>>(inputs, types, ep, out);
    } else {
        const size_t shmem = (size_t)(2 * T_J * T_K + T_J) * sizeof(float)
                           + (size_t)T_J * sizeof(int);
        uniemb_fallback<<<NBLOCKS, 256, shmem, stream>>>(inputs, types, phase, emb, out);
    }
}